// GraphConvBlock_88785563943645
// MI455X (gfx1250) — compile-verified
//
#include <hip/hip_runtime.h>
#include <hip/hip_bf16.h>

typedef __attribute__((ext_vector_type(2))) float v2f;
typedef __attribute__((ext_vector_type(8))) float v8f;

#define CH 64
#define BN_EPS 1e-5f
#define WS2_STRIDE 80   // float2 row stride: 160 dwords = 32 banks shift between K-pair rows

// ---------------------------------------------------------------- init
__global__ void gcn_init_kernel(float* __restrict__ deg, float* __restrict__ gsum,
                                float* __restrict__ gsumsq, int N) {
    int i = blockIdx.x * blockDim.x + threadIdx.x;
    if (i < N) deg[i] = 1.0f;                 // self-loop contributes 1 to degree
    if (i < CH) { gsum[i] = 0.0f; gsumsq[i] = 0.0f; }
}

// ---------------------------------------------------------------- degree
__global__ void gcn_deg_kernel(const int* __restrict__ dst, float* __restrict__ deg, int E) {
    int e = blockIdx.x * blockDim.x + threadIdx.x;
    if (e < E) atomicAdd(&deg[dst[e]], 1.0f);
}

__global__ void gcn_dinv_kernel(float* __restrict__ deg, int N) {
    int i = blockIdx.x * blockDim.x + threadIdx.x;
    if (i < N) {
        float d = deg[i];
        deg[i] = (d > 0.0f) ? rsqrtf(d) : 0.0f;   // in-place: deg -> dinv
    }
}

// ---------------------------------------------------------------- GEMM h = x @ W  (WMMA f32 16x16x4)
// One wave per 16-row slab, 4 column tiles. W staged in LDS as K-pair float2s so
// each B fragment is one ds_load_b64. Out-of-range A rows are CLAMPED (not zeroed):
// A row m only contributes to C row m, and those stores are guarded.
__global__ __launch_bounds__(256) void gcn_gemm_wmma_kernel(const float* __restrict__ x,
                                                            const float* __restrict__ W,
                                                            float* __restrict__ h, int N) {
    __shared__ float2 Ws2[32 * WS2_STRIDE];   // Ws2[k/2][col] = {W[k][col], W[k+1][col]}
    {
        float* wsf = (float*)Ws2;
        for (int i = threadIdx.x; i < 64 * 64; i += 256) {
            int k = i >> 6, col = i & 63;
            wsf[((k >> 1) * WS2_STRIDE + col) * 2 + (k & 1)] = W[i];
        }
    }
    __syncthreads();

    const int wave = threadIdx.x >> 5;
    const int lane = threadIdx.x & 31;
    const int half = lane >> 4;      // 0: lanes 0-15, 1: lanes 16-31
    const int lm   = lane & 15;
    const int row0 = (blockIdx.x * 8 + wave) * 16;
    if (row0 >= N) return;

    v8f acc0 = {}, acc1 = {}, acc2 = {}, acc3 = {};

    int arow = row0 + lm;
    if (arow >= N) arow = N - 1;                  // clamp: affected C rows never stored
    const float2* xrow2 = (const float2*)(x + (size_t)arow * CH);

    #pragma unroll
    for (int k0 = 0; k0 < 64; k0 += 4) {
        const int kp = (k0 >> 1) + half;          // K-pair owned by this half-wave
        float2 av = xrow2[kp];                    // one global_load_b64
        v2f a; a.x = av.x; a.y = av.y;

        const float2* wrow = &Ws2[kp * WS2_STRIDE];
        float2 b0v = wrow[ 0 + lm];
        float2 b1v = wrow[16 + lm];
        float2 b2v = wrow[32 + lm];
        float2 b3v = wrow[48 + lm];
        v2f b0; b0.x = b0v.x; b0.y = b0v.y;
        v2f b1; b1.x = b1v.x; b1.y = b1v.y;
        v2f b2; b2.x = b2v.x; b2.y = b2v.y;
        v2f b3; b3.x = b3v.x; b3.y = b3v.y;

        acc0 = __builtin_amdgcn_wmma_f32_16x16x4_f32(false, a, false, b0, (short)0, acc0, false, false);
        acc1 = __builtin_amdgcn_wmma_f32_16x16x4_f32(false, a, false, b1, (short)0, acc1, false, false);
        acc2 = __builtin_amdgcn_wmma_f32_16x16x4_f32(false, a, false, b2, (short)0, acc2, false, false);
        acc3 = __builtin_amdgcn_wmma_f32_16x16x4_f32(false, a, false, b3, (short)0, acc3, false, false);
    }

    // C/D layout: VGPR r -> M = r + 8*half, N = lm
    #pragma unroll
    for (int r = 0; r < 8; ++r) {
        const int orow = row0 + r + 8 * half;
        if (orow < N) {
            float* hr = h + (size_t)orow * CH;
            hr[ 0 + lm] = acc0[r];
            hr[16 + lm] = acc1[r];
            hr[32 + lm] = acc2[r];
            hr[48 + lm] = acc3[r];
        }
    }
}

// ---------------------------------------------------------------- self-loop init: out = dinv^2 * h
__global__ void gcn_selfloop_kernel(const float4* __restrict__ h, const float* __restrict__ dinv,
                                    float4* __restrict__ out, int N) {
    long long i = (long long)blockIdx.x * blockDim.x + threadIdx.x;   // over N*16 float4s
    if (i < (long long)N * (CH / 4)) {
        int n = (int)(i >> 4);
        float dv = dinv[n];
        float s = dv * dv;
        float4 v = h[i];
        v.x *= s; v.y *= s; v.z *= s; v.w *= s;
        out[i] = v;
    }
}

// ---------------------------------------------------------------- edge scatter-add
// One wave per edge: edge metadata forced wave-uniform -> scalar (SMEM) loads;
// each lane handles 2 channels: one coalesced b64 load of h, two f32 atomics.
__global__ __launch_bounds__(256) void gcn_scatter_kernel(const int* __restrict__ src,
                                                          const int* __restrict__ dst,
                                                          const float* __restrict__ dinv,
                                                          const float* __restrict__ h,
                                                          float* __restrict__ out, int E) {
    const int wave = threadIdx.x >> 5;
    const int lane = threadIdx.x & 31;
    const int e = __builtin_amdgcn_readfirstlane(blockIdx.x * 8 + wave);
    if (e >= E) return;
    const int s = src[e];                       // scalar loads (uniform address)
    const int d = dst[e];
    const float nrm = dinv[s] * dinv[d];
    const float2 hv = ((const float2*)(h + (size_t)s * CH))[lane];
    float* op = out + (size_t)d * CH + lane * 2;
    atomicAdd(op,     nrm * hv.x);
    atomicAdd(op + 1, nrm * hv.y);
}

// ---------------------------------------------------------------- per-channel sum / sumsq
__global__ __launch_bounds__(256) void gcn_stats_kernel(const float* __restrict__ out,
                                                        float* __restrict__ gsum,
                                                        float* __restrict__ gsumsq, int N) {
    __shared__ float ss[256];
    __shared__ float ss2[256];
    const int c   = threadIdx.x & 63;
    const int grp = threadIdx.x >> 6;   // 4 row-groups per block
    float s = 0.0f, s2 = 0.0f;
    for (long long r = (long long)blockIdx.x * 4 + grp; r < N; r += (long long)gridDim.x * 4) {
        float v = out[r * CH + c];
        s += v; s2 += v * v;
    }
    ss[threadIdx.x] = s; ss2[threadIdx.x] = s2;
    __syncthreads();
    if (threadIdx.x < 128) {
        ss[threadIdx.x]  += ss[threadIdx.x + 128];
        ss2[threadIdx.x] += ss2[threadIdx.x + 128];
    }
    __syncthreads();
    if (threadIdx.x < 64) {
        atomicAdd(&gsum[threadIdx.x],   ss[threadIdx.x]  + ss[threadIdx.x + 64]);
        atomicAdd(&gsumsq[threadIdx.x], ss2[threadIdx.x] + ss2[threadIdx.x + 64]);
    }
}

// ---------------------------------------------------------------- batch-norm + ReLU (in place)
// Reference bias cancels exactly under batch-stat normalization -> omitted.
__global__ void gcn_bn_relu_kernel(float4* __restrict__ out, const float* __restrict__ gsum,
                                   const float* __restrict__ gsumsq,
                                   const float* __restrict__ gamma, const float* __restrict__ beta,
                                   int N) {
    long long i = (long long)blockIdx.x * blockDim.x + threadIdx.x;   // over N*16 float4s
    if (i < (long long)N * (CH / 4)) {
        const int c0 = (int)(i & 15) * 4;
        const float invN = 1.0f / (float)N;
        float4 v = out[i];
        float* vp = &v.x;
        #pragma unroll
        for (int j = 0; j < 4; ++j) {
            int c = c0 + j;
            float mean = gsum[c] * invN;
            float var  = gsumsq[c] * invN - mean * mean;
            float y = (vp[j] - mean) * rsqrtf(var + BN_EPS) * gamma[c] + beta[c];
            vp[j] = y > 0.0f ? y : 0.0f;
        }
        out[i] = v;
    }
}

// ---------------------------------------------------------------- launch
extern "C" void kernel_launch(void* const* d_in, const int* in_sizes, int n_in,
                              void* d_out, int out_size, void* d_ws, size_t ws_size,
                              hipStream_t stream) {
    const int N = in_sizes[0] / CH;
    const int E = in_sizes[1] / 2;

    const float* x     = (const float*)d_in[0];
    const int*   ei    = (const int*)d_in[1];
    const float* W     = (const float*)d_in[2];
    // d_in[3] = bias: cancels under batch-norm, unused.
    const float* gamma = (const float*)d_in[4];
    const float* beta  = (const float*)d_in[5];
    const int* src = ei;
    const int* dst = ei + E;

    float* out    = (float*)d_out;
    float* dinv   = (float*)d_ws;                 // N floats (degree, then dinv in place)
    float* h      = dinv + N;                     // N*64 floats
    float* gsum   = h + (size_t)N * CH;           // 64 floats
    float* gsumsq = gsum + CH;                    // 64 floats

    const int B = 256;

    gcn_init_kernel<<<(N + B - 1) / B, B, 0, stream>>>(dinv, gsum, gsumsq, N);
    gcn_deg_kernel<<<(E + B - 1) / B, B, 0, stream>>>(dst, dinv, E);
    gcn_dinv_kernel<<<(N + B - 1) / B, B, 0, stream>>>(dinv, N);

    gcn_gemm_wmma_kernel<<<(N + 127) / 128, B, 0, stream>>>(x, W, h, N);

    long long nv4 = (long long)N * (CH / 4);
    gcn_selfloop_kernel<<<(unsigned)((nv4 + B - 1) / B), B, 0, stream>>>(
        (const float4*)h, dinv, (float4*)out, N);

    gcn_scatter_kernel<<<(E + 7) / 8, B, 0, stream>>>(src, dst, dinv, h, out, E);

    gcn_stats_kernel<<<1024, B, 0, stream>>>(out, gsum, gsumsq, N);
    gcn_bn_relu_kernel<<<(unsigned)((nv4 + B - 1) / B), B, 0, stream>>>(
        (float4*)out, gsum, gsumsq, gamma, beta, N);
}